// MemoryAccess_75660143886737
// MI455X (gfx1250) — compile-verified
//
#include <hip/hip_runtime.h>
#include <hip/hip_bf16.h>

// ---------------------------------------------------------------------------
// DNC MemoryAccess for MI455X (gfx1250).
//   B=16, T=64, I=1024, N=512, W=64, R=4, NW=1, TOT=471 (padded to 480)
// Kernel 1: H = X @ K + bias with V_WMMA_F32_16X16X4_F32 (exact fp32 WMMA),
//           branchless inner loop (EXEC stays all-ones, no saveexec churn).
// Kernel 2: per-batch sequential scan, memory matrix resident in LDS
//           (stride 65 to avoid bank conflicts), link matrix in L2.
// ---------------------------------------------------------------------------

typedef float v2f __attribute__((ext_vector_type(2)));
typedef float v8f __attribute__((ext_vector_type(8)));

#define NB   16
#define TT   64
#define II   1024
#define NN   512
#define WW   64
#define RR   4
#define TOTR 471          // real param width
#define TOTP 480          // padded to 30 * 16
#define MSTR 65           // LDS row stride for memory matrix (bank-conflict free)
#define EPSF 1e-6f

// ---------------------------------------------------------------------------
// Kernel 1: WMMA GEMM  H[1024 x 480] = X[1024 x 1024] @ K[1024 x 471] + bias
// One wave (32 threads) per 16x16 output tile. K-loop step 4 (f32 WMMA).
// Out-of-range columns handled by clamped address + 0/1 mask (hoisted), so
// the inner loop is branchless and EXEC remains all-ones for WMMA.
// ---------------------------------------------------------------------------
__global__ __launch_bounds__(32) void gemm_wmma_f32(
    const float* __restrict__ X, const float* __restrict__ Kmat,
    const float* __restrict__ bias, float* __restrict__ H)
{
    const int tile = blockIdx.x;
    const int ct   = tile % (TOTP / 16);   // 0..29
    const int rt   = tile / (TOTP / 16);   // 0..63
    const int lane = threadIdx.x;
    const int half = lane >> 4;            // 0: K+0/K+1 ; 1: K+2/K+3
    const int l16  = lane & 15;

    const float* Xrow = X + (size_t)(rt * 16 + l16) * II;
    const int   c     = ct * 16 + l16;
    const int   cc    = (c < TOTR) ? c : (TOTR - 1);   // always in-bounds
    const float bmask = (c < TOTR) ? 1.0f : 0.0f;      // zero out pad columns
    const float* Bcol = Kmat + cc;

    v8f acc = {};
#pragma unroll 4
    for (int k = 0; k < II; k += 4) {
        // A fragment: 16x4 f32, VGPR0 = K+2*half, VGPR1 = K+2*half+1
        v2f a;
        a.x = Xrow[k + 2 * half + 0];
        a.y = Xrow[k + 2 * half + 1];
        // B fragment: 4x16 f32, mirrored K layout, column = lane&15
        v2f bf;
        bf.x = Bcol[(size_t)(k + 2 * half + 0) * TOTR] * bmask;
        bf.y = Bcol[(size_t)(k + 2 * half + 1) * TOTR] * bmask;
        acc = __builtin_amdgcn_wmma_f32_16x16x4_f32(
            /*neg_a=*/false, a, /*neg_b=*/false, bf,
            /*c_mod=*/(short)0, acc, /*reuse_a=*/false, /*reuse_b=*/false);
    }

    const float bb = bias[cc] * bmask;
#pragma unroll
    for (int r = 0; r < 8; ++r) {
        const int m = r + half * 8;        // C/D layout: lanes 16-31 hold M+8
        H[(size_t)(rt * 16 + m) * TOTP + c] = acc[r] + bb;
    }
}

// ---------------------------------------------------------------------------
// Device helpers (scan kernel)
// ---------------------------------------------------------------------------
__device__ __forceinline__ float sigm(float x) {
    return 1.0f / (1.0f + __expf(-x));
}
__device__ __forceinline__ float softplus(float x) {
    return (x > 20.0f) ? x : log1pf(__expf(x));
}

// Block-wide softmax over 512 values (one per thread). Ends with a barrier so
// `red` can be reused immediately afterwards.
__device__ __forceinline__ float block_softmax512(float v, float* red, int n) {
    red[n] = v; __syncthreads();
#pragma unroll
    for (int off = 256; off > 0; off >>= 1) {
        if (n < off) red[n] = fmaxf(red[n], red[n + off]);
        __syncthreads();
    }
    const float mx = red[0]; __syncthreads();
    const float e = __expf(v - mx);
    red[n] = e; __syncthreads();
#pragma unroll
    for (int off = 256; off > 0; off >>= 1) {
        if (n < off) red[n] += red[n + off];
        __syncthreads();
    }
    const float s = red[0]; __syncthreads();
    return e / s;
}

__device__ __forceinline__ float block_sum512(float v, float* red, int n) {
    red[n] = v; __syncthreads();
#pragma unroll
    for (int off = 256; off > 0; off >>= 1) {
        if (n < off) red[n] += red[n + off];
        __syncthreads();
    }
    const float s = red[0]; __syncthreads();
    return s;
}

// ---------------------------------------------------------------------------
// Kernel 2: sequential DNC scan. grid = 16 (one batch per WGP), block = 512.
// Dynamic LDS layout (floats):
//   mem[512*65] | h[480] | ww[512] | wc[512] | usg[512] | prc[512]
//   | red[512] | srt[512] | rc[4*512] | rwp[4*512] | irank[512 ints]
// Total = 41440 words = 165760 bytes  (<< 320 KB WGP LDS).
// ---------------------------------------------------------------------------
__global__ __launch_bounds__(512) void dnc_scan(
    const float* __restrict__ H,
    const float* __restrict__ mem0, const float* __restrict__ usage0,
    const float* __restrict__ prec0, const float* __restrict__ rw0,
    float* __restrict__ link,
    float* __restrict__ out_rw, float* __restrict__ out_mem,
    float* __restrict__ out_usage)
{
    extern __shared__ float sm[];
    float* mem  = sm;                       // 512*65
    float* h    = mem  + NN * MSTR;         // 480
    float* ww   = h    + TOTP;              // 512
    float* wc   = ww   + NN;                // 512
    float* usg  = wc   + NN;                // 512
    float* prc  = usg  + NN;                // 512
    float* red  = prc  + NN;                // 512
    float* srt  = red  + NN;                // 512
    float* rc   = srt  + NN;                // 4*512
    float* rwp  = rc   + RR * NN;           // 4*512
    int*  irank = (int*)(rwp + RR * NN);    // 512

    const int b = blockIdx.x;
    const int n = threadIdx.x;
    float* LK = link + (size_t)b * NN * NN;

    // ---- initial state ----
    for (int i = n; i < NN * WW; i += NN)
        mem[(i >> 6) * MSTR + (i & 63)] = mem0[(size_t)b * NN * WW + i];
    usg[n] = usage0[b * NN + n];
    prc[n] = prec0[b * NN + n];                       // NW==1
#pragma unroll
    for (int r = 0; r < RR; ++r)
        rwp[r * NN + n] = rw0[((size_t)b * RR + r) * NN + n];
    __syncthreads();

    for (int t = 0; t < TT; ++t) {
        // ---- fetch h_t ----
        const float* hrow = H + (size_t)(b * TT + t) * TOTP;
        for (int i = n; i < TOTR; i += NN) h[i] = hrow[i];
        __syncthreads();

        // ---- per-thread scalar params ----
        const float wg = sigm(h[128]);
        const float ag = sigm(h[129]);
        float fg[RR];
#pragma unroll
        for (int r = 0; r < RR; ++r) fg[r] = sigm(h[130 + r]);
        float rm[RR][3];
#pragma unroll
        for (int r = 0; r < RR; ++r) {
            const float a0 = h[134 + r * 3], a1 = h[135 + r * 3], a2 = h[136 + r * 3];
            const float mx = fmaxf(a0, fmaxf(a1, a2));
            const float e0 = __expf(a0 - mx), e1 = __expf(a1 - mx), e2 = __expf(a2 - mx);
            const float s = e0 + e1 + e2;
            rm[r][0] = e0 / s; rm[r][1] = e1 / s; rm[r][2] = e2 / s;
        }
        const float wstr = softplus(h[210]);
        float rstr[RR];
#pragma unroll
        for (int r = 0; r < RR; ++r) rstr[r] = softplus(h[467 + r]);

        // ---- memory row norm (old memory) ----
        float mn2 = 0.f;
#pragma unroll 8
        for (int w = 0; w < WW; ++w) { const float m = mem[n * MSTR + w]; mn2 += m * m; }
        const float mn = sqrtf(mn2);

        // ---- write-content weights ----
        {
            float kn2 = 0.f, dot = 0.f;
#pragma unroll 8
            for (int w = 0; w < WW; ++w) {
                const float kv = h[146 + w];
                kn2 += kv * kv;
                dot += kv * mem[n * MSTR + w];
            }
            const float sim = dot / (sqrtf(kn2) * mn + EPSF) * wstr;
            wc[n] = block_softmax512(sim, red, n);
        }
        // ---- read-content weights ----
        for (int r = 0; r < RR; ++r) {
            float kn2 = 0.f, dot = 0.f;
#pragma unroll 8
            for (int w = 0; w < WW; ++w) {
                const float kv = h[211 + r * WW + w];
                kn2 += kv * kv;
                dot += kv * mem[n * MSTR + w];
            }
            const float sim = dot / (sqrtf(kn2) * mn + EPSF) * rstr[r];
            rc[r * NN + n] = block_softmax512(sim, red, n);
        }

        // ---- allocation: rank sort + product scan over sorted usage ----
        const float u = usg[n];
        int rk = 0;
        for (int m = 0; m < NN; ++m) {
            const float um = usg[m];
            rk += (um < u) || (um == u && m < n);
        }
        irank[n] = rk;
        srt[rk] = u;
        __syncthreads();
        const float su_pos = srt[n];            // sorted usage at position n
#pragma unroll
        for (int off = 1; off < NN; off <<= 1) { // inclusive product scan
            const float a  = srt[n];
            const float ap = (n >= off) ? srt[n - off] : 1.0f;
            __syncthreads();
            srt[n] = a * ap;
            __syncthreads();
        }
        const float pex = (n == 0) ? 1.0f : srt[n - 1];
        red[n] = (1.0f - su_pos) * pex;          // sa at sorted position n
        __syncthreads();
        const float al = red[irank[n]];
        __syncthreads();

        // ---- write weights ----
        const float wwn = wg * (ag * al + (1.0f - ag) * wc[n]);
        ww[n] = wwn;
        const float sww = block_sum512(wwn, red, n);

        // ---- memory erase + write (new memory) ----
#pragma unroll 8
        for (int w = 0; w < WW; ++w) {
            const float ev = sigm(h[64 + w]);
            const float wv = h[w];
            const float mv = mem[n * MSTR + w];
            mem[n * MSTR + w] = mv * (1.0f - wwn * ev) + wwn * wv;
        }

        // ---- usage update (uses OLD rwp) ----
        {
            const float uw = u + wwn - u * wwn;
            float psi = 1.0f;
#pragma unroll
            for (int r = 0; r < RR; ++r) psi *= (1.0f - fg[r] * rwp[r * NN + n]);
            usg[n] = uw * psi;
        }
        __syncthreads();   // ww[], new mem[], visible to everyone

        // ---- link pass 1: row-wise update + fwd (registers) ----
        float f0 = 0.f, f1 = 0.f, f2 = 0.f, f3 = 0.f;
        {
            float4* lrow = (float4*)(LK + (size_t)n * NN);
            for (int m4 = 0; m4 < NN / 4; ++m4) {
                float4 lk = lrow[m4];
                const int m = m4 * 4;
                float lv[4] = {lk.x, lk.y, lk.z, lk.w};
#pragma unroll
                for (int j = 0; j < 4; ++j) {
                    const int mm = m + j;
                    float ln = (1.0f - wwn - ww[mm]) * lv[j] + wwn * prc[mm];
                    if (mm == n) ln = 0.0f;
                    lv[j] = ln;
                    f0 += ln * rwp[0 * NN + mm];
                    f1 += ln * rwp[1 * NN + mm];
                    f2 += ln * rwp[2 * NN + mm];
                    f3 += ln * rwp[3 * NN + mm];
                }
                lk.x = lv[0]; lk.y = lv[1]; lk.z = lv[2]; lk.w = lv[3];
                lrow[m4] = lk;
            }
        }
        __threadfence();      // link_new visible across the workgroup (L2)
        __syncthreads();

        // ---- link pass 2: column-wise bwd (registers) ----
        float b0 = 0.f, b1 = 0.f, b2 = 0.f, b3 = 0.f;
        for (int m = 0; m < NN; ++m) {
            const float lk = LK[(size_t)m * NN + n];
            b0 += lk * rwp[0 * NN + m];
            b1 += lk * rwp[1 * NN + m];
            b2 += lk * rwp[2 * NN + m];
            b3 += lk * rwp[3 * NN + m];
        }

        // ---- precedence update (own element; pass1 already consumed prc) ----
        prc[n] = (1.0f - sww) * prc[n] + wwn;

        // ---- new read weights ----
        const float r0 = rm[0][0] * rc[0 * NN + n] + rm[0][1] * f0 + rm[0][2] * b0;
        const float r1 = rm[1][0] * rc[1 * NN + n] + rm[1][1] * f1 + rm[1][2] * b1;
        const float r2 = rm[2][0] * rc[2 * NN + n] + rm[2][1] * f2 + rm[2][2] * b2;
        const float r3 = rm[3][0] * rc[3 * NN + n] + rm[3][1] * f3 + rm[3][2] * b3;
        __syncthreads();                 // everyone done reading old rwp
        rwp[0 * NN + n] = r0;
        rwp[1 * NN + n] = r1;
        rwp[2 * NN + n] = r2;
        rwp[3 * NN + n] = r3;
        __syncthreads();

        // ---- read words: out[b][t][r][w] = sum_n rw[r][n] * mem_new[n][w] ----
        if (n < RR * WW) {
            const int r = n >> 6, w = n & 63;
            float acc = 0.f;
            for (int m = 0; m < NN; ++m)
                acc += rwp[r * NN + m] * mem[m * MSTR + w];
            out_rw[(((size_t)b * TT + t) * RR + r) * WW + w] = acc;
        }
        __syncthreads();
    }

    // ---- final memory + usage outputs ----
    for (int i = n; i < NN * WW; i += NN)
        out_mem[(size_t)b * NN * WW + i] = mem[(i >> 6) * MSTR + (i & 63)];
    out_usage[b * NN + n] = usg[n];
}

// ---------------------------------------------------------------------------
// Host launcher
// ---------------------------------------------------------------------------
extern "C" void kernel_launch(void* const* d_in, const int* in_sizes, int n_in,
                              void* d_out, int out_size, void* d_ws, size_t ws_size,
                              hipStream_t stream)
{
    const float* x      = (const float*)d_in[0];   // (16,64,1024)
    const float* kmat   = (const float*)d_in[1];   // (1024,471)
    const float* bias   = (const float*)d_in[2];   // (471)
    const float* mem0   = (const float*)d_in[3];   // (16,512,64)
    const float* usage0 = (const float*)d_in[4];   // (16,512)
    const float* link0  = (const float*)d_in[5];   // (16,1,512,512)
    const float* prec0  = (const float*)d_in[6];   // (16,1,512)
    const float* rw0    = (const float*)d_in[7];   // (16,4,512)

    // workspace layout
    float* H    = (float*)d_ws;                               // 1024*480 f32
    float* link = H + (size_t)(NB * TT) * TOTP;               // 16*512*512 f32

    // link state starts as link0 (device-to-device, capture-safe)
    hipMemcpyAsync(link, link0, (size_t)NB * NN * NN * sizeof(float),
                   hipMemcpyDeviceToDevice, stream);

    // 1) projection GEMM via f32 WMMA
    gemm_wmma_f32<<<(NB * TT / 16) * (TOTP / 16), 32, 0, stream>>>(x, kmat, bias, H);

    // 2) sequential scan, one batch per workgroup
    const size_t smem = (size_t)(NN * MSTR + TOTP + 6 * NN + 2 * RR * NN + NN)
                        * sizeof(float);
    hipFuncSetAttribute((const void*)dnc_scan,
                        hipFuncAttributeMaxDynamicSharedMemorySize, (int)smem);

    float* out_rw    = (float*)d_out;                          // (16,64,4,64)
    float* out_mem   = out_rw + (size_t)NB * TT * RR * WW;     // (16,512,64)
    float* out_usage = out_mem + (size_t)NB * NN * WW;         // (16,512)

    dnc_scan<<<NB, NN, smem, stream>>>(H, mem0, usage0, prec0, rw0, link,
                                       out_rw, out_mem, out_usage);
    (void)in_sizes; (void)n_in; (void)out_size; (void)ws_size;
}